// Identity_79173427135079
// MI455X (gfx1250) — compile-verified
//
#include <hip/hip_runtime.h>

// Sparse 2-layer COO linear for MI455X (gfx1250, wave32).
// Strategy: build CSR on device (histogram/scan/scatter), run each layer as a
// gather-based row kernel (no float atomics), stage per-row edge lists into LDS
// with CDNA5 async-to-LDS copies (ASYNCcnt), broadcast edges with v_readlane,
// scalar-addressed 256B batch gathers. WMMA is structurally inapplicable here
// (random sparsity has no shared-K tile structure); the op is memory-bound.

typedef unsigned int u32;
typedef unsigned long long u64;

#define ROWS_PER_BLOCK 8
#define EDGE_CAP 512   // per-wave LDS staging capacity (int2 = 8B) -> 4KB/wave, 32KB/block

__global__ void zero_u32_kernel(u32* __restrict__ p, int n) {
  int i = blockIdx.x * blockDim.x + threadIdx.x;
  if (i < n) p[i] = 0u;
}

__global__ void hist_kernel(const int* __restrict__ row, u32* __restrict__ counts, int nnz) {
  int e = blockIdx.x * blockDim.x + threadIdx.x;
  if (e < nnz) atomicAdd(&counts[row[e]], 1u);
}

// Single-block exclusive scan over `dim` counts; writes offsets and a working cursor copy.
__global__ __launch_bounds__(1024) void scan_kernel(const u32* __restrict__ counts,
                                                    u32* __restrict__ offsets,
                                                    u32* __restrict__ cursor, int dim) {
  __shared__ u32 sums[1024];
  int t = threadIdx.x;
  int per = (dim + 1023) >> 10;
  int s = t * per;
  u32 local = 0;
  for (int i = 0; i < per; ++i) { int idx = s + i; if (idx < dim) local += counts[idx]; }
  sums[t] = local;
  __syncthreads();
  for (int off = 1; off < 1024; off <<= 1) {      // Hillis-Steele inclusive scan
    u32 v = 0;
    if (t >= off) v = sums[t - off];
    __syncthreads();
    if (t >= off) sums[t] += v;
    __syncthreads();
  }
  u32 base = (t == 0) ? 0u : sums[t - 1];
  for (int i = 0; i < per; ++i) {
    int idx = s + i;
    if (idx < dim) { offsets[idx] = base; cursor[idx] = base; base += counts[idx]; }
  }
}

__global__ void build_kernel(const int* __restrict__ row, const int* __restrict__ col,
                             const float* __restrict__ w1, const float* __restrict__ w2,
                             u32* __restrict__ cursor, int2* __restrict__ e1,
                             int2* __restrict__ e2, int nnz) {
  int e = blockIdx.x * blockDim.x + threadIdx.x;
  if (e >= nnz) return;
  int r = row[e];
  u32 p = atomicAdd(&cursor[r], 1u);
  int c = col[e];
  e1[p] = make_int2(c, __float_as_int(w1[e]));
  e2[p] = make_int2(c, __float_as_int(w2[e]));
}

// x[64][dim] -> xT[dim][64], 64x64 tiles via LDS (65-pad: conflict-free both phases)
__global__ __launch_bounds__(256) void transpose_in_kernel(const float* __restrict__ x,
                                                           float* __restrict__ xT, int dim) {
  __shared__ float tile[64][65];
  int d0 = blockIdx.x * 64;
  int tx = threadIdx.x & 63;
  int tq = threadIdx.x >> 6;
#pragma unroll
  for (int i = 0; i < 16; ++i) {
    int b = tq * 16 + i;
    tile[tx][b] = x[(size_t)b * dim + (d0 + tx)];  // coalesced over tx (d)
  }
  __syncthreads();
#pragma unroll
  for (int i = 0; i < 16; ++i) {
    int dy = tq * 16 + i;
    xT[(size_t)(d0 + dy) * 64 + tx] = tile[dy][tx];  // coalesced over tx (b)
  }
}

// yT[dim][64] -> y[64][dim]
__global__ __launch_bounds__(256) void transpose_out_kernel(const float* __restrict__ yT,
                                                            float* __restrict__ y, int dim) {
  __shared__ float tile[64][65];
  int d0 = blockIdx.x * 64;
  int tx = threadIdx.x & 63;
  int tq = threadIdx.x >> 6;
#pragma unroll
  for (int i = 0; i < 16; ++i) {
    int dy = tq * 16 + i;
    tile[dy][tx] = yT[(size_t)(d0 + dy) * 64 + tx];  // coalesced over tx (b)
  }
  __syncthreads();
#pragma unroll
  for (int i = 0; i < 16; ++i) {
    int b = tq * 16 + i;
    y[(size_t)b * dim + (d0 + tx)] = tile[tx][b];    // coalesced over tx (d)
  }
}

// One wave32 per output row. Lane L owns batch elements {2L, 2L+1} (B=64).
// Row's edge list is async-staged into LDS (one burst, max memory-level
// parallelism), then consumed in groups of 32: one coalesced ds_load_b64 per
// group, broadcast per-edge via v_readlane (immediate lane index). Two FMA
// accumulator chains hide VALU latency.
__global__ __launch_bounds__(256) void spmv_kernel(const int2* __restrict__ edges,
                                                   const u32* __restrict__ offsets,
                                                   const u32* __restrict__ counts,
                                                   const float* __restrict__ inT,
                                                   const float* __restrict__ bias,
                                                   float* __restrict__ outT, int dim) {
  __shared__ int2 smE[ROWS_PER_BLOCK][EDGE_CAP];
  int wave = threadIdx.x >> 5;
  int lane = threadIdx.x & 31;
  int r = __builtin_amdgcn_readfirstlane((int)(blockIdx.x * ROWS_PER_BLOCK + wave));
  if (r >= dim) return;
  u32 start = (u32)__builtin_amdgcn_readfirstlane((int)offsets[r]);
  u32 cnt   = (u32)__builtin_amdgcn_readfirstlane((int)counts[r]);
  int2* sm = smE[wave];
  const int2* gsrc = edges + start;
  u32 stage = cnt < (u32)EDGE_CAP ? cnt : (u32)EDGE_CAP;

  // CDNA5 async copy: global -> LDS, tracked by ASYNCcnt (8B per lane per issue).
  for (u32 i = (u32)lane; i < stage; i += 32u) {
    unsigned ldsAddr = (unsigned)(u64)(&sm[i]);  // low 32 bits of generic ptr = LDS offset
    unsigned voff = i * 8u;
    asm volatile("global_load_async_to_lds_b64 %0, %1, %2"
                 :: "v"(ldsAddr), "v"(voff), "s"((u64)gsrc)
                 : "memory");
  }

  float bv = bias ? bias[r] : 0.0f;
  float2 acc0, acc1;
  acc0.x = bv;  acc0.y = bv;    // even-edge chain (carries bias)
  acc1.x = 0.f; acc1.y = 0.f;   // odd-edge chain
  const float2* xrb = (const float2*)inT;

  __builtin_amdgcn_s_wait_asynccnt(0);  // per-wave: our LDS chunk is written

  u32 nfull = stage >> 5;         // complete groups of 32 edges
  for (u32 g = 0; g < nfull; ++g) {
    int2 ev = sm[g * 32 + lane];  // one coalesced LDS read per group
#pragma unroll
    for (int j = 0; j < 32; ++j) {
      int c = __builtin_amdgcn_readlane(ev.x, j);
      float w = __int_as_float(__builtin_amdgcn_readlane(ev.y, j));
      float2 v = xrb[(size_t)c * 32 + lane];  // 256B contiguous gather (L2-resident)
      if (j & 1) {
        acc1.x = fmaf(w, v.x, acc1.x);
        acc1.y = fmaf(w, v.y, acc1.y);
      } else {
        acc0.x = fmaf(w, v.x, acc0.x);
        acc0.y = fmaf(w, v.y, acc0.y);
      }
    }
  }
  u32 rem = stage - (nfull << 5);
  if (rem) {
    int2 ev = sm[nfull * 32 + lane];  // in-bounds (cap 512); unstaged lanes unused
#pragma unroll
    for (int j = 0; j < 31; ++j) {
      if ((u32)j < rem) {
        int c = __builtin_amdgcn_readlane(ev.x, j);
        float w = __int_as_float(__builtin_amdgcn_readlane(ev.y, j));
        float2 v = xrb[(size_t)c * 32 + lane];
        if (j & 1) {
          acc1.x = fmaf(w, v.x, acc1.x);
          acc1.y = fmaf(w, v.y, acc1.y);
        } else {
          acc0.x = fmaf(w, v.x, acc0.x);
          acc0.y = fmaf(w, v.y, acc0.y);
        }
      }
    }
  }
  // Overflow fallback (statistically never hit for mean 64 edges/row, cap 512)
  for (u32 k = stage; k < cnt; ++k) {
    int2 e = gsrc[k];
    int c = __builtin_amdgcn_readfirstlane(e.x);
    float w = __int_as_float(__builtin_amdgcn_readfirstlane(e.y));
    float2 v = xrb[(size_t)c * 32 + lane];
    acc0.x = fmaf(w, v.x, acc0.x);
    acc0.y = fmaf(w, v.y, acc0.y);
  }

  float2 acc;
  acc.x = acc0.x + acc1.x;
  acc.y = acc0.y + acc1.y;
  ((float2*)outT)[(size_t)r * 32 + lane] = acc;  // single coalesced 256B row store
}

extern "C" void kernel_launch(void* const* d_in, const int* in_sizes, int n_in,
                              void* d_out, int out_size, void* d_ws, size_t ws_size,
                              hipStream_t stream) {
  const float* x  = (const float*)d_in[0];
  const float* w1 = (const float*)d_in[1];
  const float* b1 = (const float*)d_in[2];
  const float* w2 = (const float*)d_in[3];
  const int*   row = (const int*)d_in[4];
  const int*   col = (const int*)d_in[5];
  float* y = (float*)d_out;

  const int nnz = in_sizes[1];  // w1 length
  const int dim = in_sizes[2];  // b1 length; B = in_sizes[0]/dim == 64 by construction

  char* p = (char*)d_ws;
  auto alloc = [&](size_t bytes) -> void* {
    void* q = (void*)p;
    p += (bytes + 255) & ~(size_t)255;
    return q;
  };
  u32* counts  = (u32*)alloc((size_t)dim * 4);
  u32* offsets = (u32*)alloc((size_t)dim * 4);
  u32* cursor  = (u32*)alloc((size_t)dim * 4);
  float* xT = (float*)alloc((size_t)dim * 64 * 4);
  float* hT = (float*)alloc((size_t)dim * 64 * 4);
  float* yT = (float*)alloc((size_t)dim * 64 * 4);
  int2* e1 = (int2*)alloc((size_t)nnz * 8);
  int2* e2 = (int2*)alloc((size_t)nnz * 8);
  (void)ws_size; (void)n_in; (void)out_size;

  zero_u32_kernel<<<(dim + 255) / 256, 256, 0, stream>>>(counts, dim);
  hist_kernel<<<(nnz + 255) / 256, 256, 0, stream>>>(row, counts, nnz);
  scan_kernel<<<1, 1024, 0, stream>>>(counts, offsets, cursor, dim);
  build_kernel<<<(nnz + 255) / 256, 256, 0, stream>>>(row, col, w1, w2, cursor, e1, e2, nnz);
  transpose_in_kernel<<<(dim + 63) / 64, 256, 0, stream>>>(x, xT, dim);

  int spmvBlocks = (dim + ROWS_PER_BLOCK - 1) / ROWS_PER_BLOCK;
  // Layer 1: hT = A(w1) * xT + b1
  spmv_kernel<<<spmvBlocks, 256, 0, stream>>>(e1, offsets, counts, xT, b1, hT, dim);
  // Layer 2: yT = A(w2) * hT
  spmv_kernel<<<spmvBlocks, 256, 0, stream>>>(e2, offsets, counts, hT, nullptr, yT, dim);

  transpose_out_kernel<<<(dim + 63) / 64, 256, 0, stream>>>(yT, y, dim);
}